// GraphConvolutionLayer_Sparse_21071109554802
// MI455X (gfx1250) — compile-verified
//
#include <hip/hip_runtime.h>

// GraphConvolutionLayer (sparse): out = segment_sum(val * (X@W)[col], row) + bias
// MI455X strategy:
//   - GEMM via V_WMMA_F32_16X16X4_F32 (exact fp32 semantics, 13.1 GFLOP -> minor cost)
//   - SpMM gather/scatter is L2-resident (h=100MB, out=100MB, L2=192MB):
//     edge-parallel wave32, hardware global_atomic_add_f32 (no-return, DEV scope)
//     via inline asm; channel offsets folded into the IOFFSET immediate so a
//     single 64-bit base address VGPR pair is reused for all 8 atomics.
//   - h staged in d_ws (100000*256*4 = 102.4 MB)

typedef __attribute__((ext_vector_type(2))) float v2f;
typedef __attribute__((ext_vector_type(8))) float v8f;

#define D_DIM 256

// ---------------------------------------------------------------------------
// out[i, :] = bias[:]   (float4 vectorized)
// ---------------------------------------------------------------------------
__global__ __launch_bounds__(256) void init_out_bias(float* __restrict__ out,
                                                     const float* __restrict__ bias,
                                                     int total4) {
  int idx = blockIdx.x * blockDim.x + threadIdx.x;
  if (idx >= total4) return;
  int c4 = idx & (D_DIM / 4 - 1);           // column chunk within the row
  ((float4*)out)[idx] = ((const float4*)bias)[c4];
}

// ---------------------------------------------------------------------------
// H = X @ W  via V_WMMA_F32_16X16X4_F32.
// One wave computes a 16-row x 64-col strip: 4 accumulator tiles, A reused.
// Fragment layouts per CDNA5 ISA 7.12.2:
//   A (16x4 f32): lane<16 holds A[M=lane][K=k..k+1], lane>=16 holds K=k+2..k+3
//   B (4x16):     lane<16 holds B[K=k][N],B[K=k+1][N]; lane>=16 K=k+2,k+3
//   C/D (16x16):  VGPR v: lane<16 -> M=v, lane>=16 -> M=v+8; N = lane&15
// Guard is wave-uniform -> EXEC all-1s inside WMMA (ISA requirement).
// ---------------------------------------------------------------------------
__global__ __launch_bounds__(256) void gemm_xw(const float* __restrict__ X,
                                               const float* __restrict__ W,
                                               float* __restrict__ H,
                                               int nRowTiles) {
  const int lane  = threadIdx.x & 31;
  const int gw    = blockIdx.x * 8 + (threadIdx.x >> 5); // global wave id
  const int rowTile = gw >> 2;
  if (rowTile >= nRowTiles) return;                      // wave-uniform exit
  const int colBase = (gw & 3) * 64;
  const int laneM   = lane & 15;
  const int kk      = (lane >> 4) << 1;                  // 0 or 2
  const int hiHalf  = (lane >> 4) << 3;                  // 0 or 8 (for C/D rows)

  v8f acc0 = {}, acc1 = {}, acc2 = {}, acc3 = {};

  const float* xrow = X + (size_t)(rowTile * 16 + laneM) * D_DIM;

  #pragma unroll 4
  for (int k = 0; k < D_DIM; k += 4) {
    v2f a = *(const v2f*)(xrow + k + kk);                // global_load_b64

    const float* w0 = W + (size_t)(k + kk) * D_DIM + colBase + laneM;
    const float* w1 = w0 + D_DIM;
    v2f b0 = { w0[0],  w1[0]  };
    v2f b1 = { w0[16], w1[16] };
    v2f b2 = { w0[32], w1[32] };
    v2f b3 = { w0[48], w1[48] };

    acc0 = __builtin_amdgcn_wmma_f32_16x16x4_f32(false, a, false, b0, (short)0, acc0, false, false);
    acc1 = __builtin_amdgcn_wmma_f32_16x16x4_f32(false, a, false, b1, (short)0, acc1, false, false);
    acc2 = __builtin_amdgcn_wmma_f32_16x16x4_f32(false, a, false, b2, (short)0, acc2, false, false);
    acc3 = __builtin_amdgcn_wmma_f32_16x16x4_f32(false, a, false, b3, (short)0, acc3, false, false);
  }

  float* hp = H + (size_t)(rowTile * 16 + hiHalf) * D_DIM + colBase + laneM;
  #pragma unroll
  for (int v = 0; v < 8; ++v) {
    float* hr = hp + (size_t)v * D_DIM;
    hr[0]  = acc0[v];
    hr[16] = acc1[v];
    hr[32] = acc2[v];
    hr[48] = acc3[v];
  }
}

// ---------------------------------------------------------------------------
// Edge-parallel scatter: one wave32 per edge, each lane owns 8 channels.
//   out[row[e], :] += val[e] * H[col[e], :]
// H and out are L2-resident (100MB each < 192MB L2); hardware f32 add atomics
// resolve in the L2 atomic units (no-return -> STOREcnt, fire-and-forget).
// One base address VGPR pair + IOFFSET immediates -> no per-element addr math.
// Kernel-end S_ENDPGM performs an implicit wait-idle.
// ---------------------------------------------------------------------------
#define GATOMIC_ADD_F32(OFFMOD, VALUE)                                        \
  asm volatile("global_atomic_add_f32 %0, %1, off " OFFMOD " scope:SCOPE_DEV" \
               :                                                              \
               : "v"(op), "v"(VALUE)                                          \
               : "memory")

__global__ __launch_bounds__(256) void spmm_scatter(const int* __restrict__ arow,
                                                    const int* __restrict__ acol,
                                                    const float* __restrict__ aval,
                                                    const float* __restrict__ H,
                                                    float* __restrict__ out,
                                                    int nEdges) {
  const int lane = threadIdx.x & 31;
  const int e    = blockIdx.x * 8 + (threadIdx.x >> 5);
  if (e >= nEdges) return;

  const int   r = arow[e];
  const int   c = acol[e];
  const float v = aval[e];

  const float4* hp = (const float4*)(H + (size_t)c * D_DIM) + lane * 2;
  float4 h0 = hp[0];
  float4 h1 = hp[1];

  float* op = out + (size_t)r * D_DIM + lane * 8;
  GATOMIC_ADD_F32("",          v * h0.x);
  GATOMIC_ADD_F32("offset:4",  v * h0.y);
  GATOMIC_ADD_F32("offset:8",  v * h0.z);
  GATOMIC_ADD_F32("offset:12", v * h0.w);
  GATOMIC_ADD_F32("offset:16", v * h1.x);
  GATOMIC_ADD_F32("offset:20", v * h1.y);
  GATOMIC_ADD_F32("offset:24", v * h1.z);
  GATOMIC_ADD_F32("offset:28", v * h1.w);
}

// ---------------------------------------------------------------------------
extern "C" void kernel_launch(void* const* d_in, const int* in_sizes, int n_in,
                              void* d_out, int out_size, void* d_ws, size_t ws_size,
                              hipStream_t stream) {
  const float* X    = (const float*)d_in[0];   // features [N, 256]
  const int*   arow = (const int*)  d_in[1];   // adj_row  [E]
  const int*   acol = (const int*)  d_in[2];   // adj_col  [E]
  const float* aval = (const float*)d_in[3];   // adj_val  [E]
  const float* W    = (const float*)d_in[4];   // weight   [256, 256] row-major
  const float* bias = (const float*)d_in[5];   // bias     [1, 256]

  float* out = (float*)d_out;
  float* H   = (float*)d_ws;                   // h = X@W staging, 102.4 MB

  const int N = in_sizes[0] / D_DIM;
  const int E = in_sizes[1];

  // 1) out = bias (broadcast rows)
  int total4 = N * (D_DIM / 4);
  init_out_bias<<<(total4 + 255) / 256, 256, 0, stream>>>(out, bias, total4);

  // 2) H = X @ W   (N=100000 is a multiple of 16 -> 6250 row tiles, no tail)
  int nRowTiles = (N + 15) / 16;
  int strips    = nRowTiles * 4;               // 4 x 64-col groups
  gemm_xw<<<(strips + 7) / 8, 256, 0, stream>>>(X, W, H, nRowTiles);

  // 3) out += scatter(val * H[col])
  spmm_scatter<<<(E + 7) / 8, 256, 0, stream>>>(arow, acol, aval, H, out, E);
}